// RSV_Loss_4234837754510
// MI455X (gfx1250) — compile-verified
//
#include <hip/hip_runtime.h>
#include <math.h>

// ---------------------------------------------------------------------------
// RSV outpainting loss for MI455X (gfx1250, wave32).
//  - 21x21 Gaussian kernel is rank-1 (sqrt(outer(d,d)) = outer(sqrt d, sqrt d)),
//    so each of the 9 iterations is a separable 21-tap H-pass + fused
//    V-pass + blending on the [16,512,512] mask planes (all fp32).
//  - Conv tiles use V_WMMA_F32_16X16X4_F32: 16x16 output tile, K=36
//    (21 taps + 16-wide tile) as 9 K=4 WMMA steps; the banded weight
//    operand is pre-swizzled into a per-lane table by a device init kernel
//    (same table serves hconv-B and vconv-A: both are w[k - lane%16]).
//  - One wave per block -> tile decode is pure SGPR; interior/edge split is
//    a scalar branch; interior path is unconditional coalesced b64 loads;
//    edge path uses clamped loads * {0,1} mask (no predicated loads/spills).
//  - Working set (~64 MB) is L2-resident (192 MB); final |x-y|*rsv reduction
//    is HBM-bound (~100 MB of x,y -> ~4.4 us at 23.3 TB/s), float4-vectorized.
// ---------------------------------------------------------------------------

typedef __attribute__((ext_vector_type(2))) float v2f;
typedef __attribute__((ext_vector_type(8))) float v8f;

static constexpr int NB = 16;      // batch
static constexpr int NC = 3;       // channels of x,y
static constexpr int NH = 512;
static constexpr int NW = 512;
static constexpr int TAPS = 21;
static constexpr int RAD  = 10;    // (21-1)/2
static constexpr int KSTEPS = 9;   // 9*4 = 36 >= 16 + 20

// ---------------------------------------------------------------------------
// Device-side weight/band-table init (graph-capture safe, no H2D copies).
// band[kk][lane][0..1] = w[(kk*4 + (lane>>4)*2 + {0,1}) - (lane&15)] or 0.
// ---------------------------------------------------------------------------
__global__ __launch_bounds__(32)
void init_band(float* __restrict__ band) {
    __shared__ float w[TAPS];
    if (threadIdx.x == 0) {
        const double sigma    = 1.0 / 40.0;
        const double interval = (2.0 * sigma + 1.0) / (double)TAPS;
        const double lo = -sigma - interval / 2.0;
        const double hi =  sigma + interval / 2.0;
        double cdf[TAPS + 1];
        for (int i = 0; i <= TAPS; ++i) {
            const double xx = lo + (hi - lo) * (double)i / (double)TAPS;
            cdf[i] = 0.5 * (1.0 + erf(xx / sqrt(2.0)));
        }
        double s[TAPS], ssum = 0.0;
        for (int i = 0; i < TAPS; ++i) { s[i] = sqrt(cdf[i + 1] - cdf[i]); ssum += s[i]; }
        for (int i = 0; i < TAPS; ++i) w[i] = (float)(s[i] / ssum);
    }
    __syncthreads();
    const int lane = threadIdx.x;
    const int l16  = lane & 15;
    const int half = lane >> 4;
    for (int kk = 0; kk < KSTEPS; ++kk) {
        const int kbase = kk * 4 + half * 2;
        const int t0 = kbase     - l16;
        const int t1 = kbase + 1 - l16;
        band[kk * 64 + lane * 2 + 0] = (t0 >= 0 && t0 < TAPS) ? w[t0] : 0.0f;
        band[kk * 64 + lane * 2 + 1] = (t1 >= 0 && t1 < TAPS) ? w[t1] : 0.0f;
    }
}

// ---------------------------------------------------------------------------
// Horizontal 21-tap conv, SAME zero padding. One wave per block, one
// 16x16 tile per wave. C = A(16x36 input patch) x B(36x16 banded weights).
// ---------------------------------------------------------------------------
__global__ __launch_bounds__(32)
void hconv_wmma(const float* __restrict__ in, float* __restrict__ out,
                const float* __restrict__ band) {
    const int lane  = threadIdx.x;
    const int tile  = blockIdx.x;          // SGPR: 16384 tiles = 16 * 32 * 32
    const int plane = tile >> 10;
    const int ty    = (tile >> 5) & 31;
    const int tx    = tile & 31;
    const int r0 = ty * 16, c0 = tx * 16;

    const float* p = in  + (size_t)plane * NH * NW;
    float*       q = out + (size_t)plane * NH * NW;

    const int half = lane >> 4;
    const int l16  = lane & 15;
    const float* arowp = p + (size_t)(r0 + l16) * NW;   // A-matrix row for lane

    v8f acc = {};
    if (tx >= 1 && tx <= 30) {
        // Interior: cols c0-10 .. c0+25 all in-range; unconditional b64 loads.
#pragma unroll
        for (int kk = 0; kk < KSTEPS; ++kk) {
            const int kbase = kk * 4 + half * 2;
            const v2f a = *(const v2f*)(arowp + (c0 + kbase - RAD));   // 8B aligned
            const v2f b = *(const v2f*)(band + kk * 64 + lane * 2);
            acc = __builtin_amdgcn_wmma_f32_16x16x4_f32(
                      false, a, false, b, (short)0, acc, false, false);
        }
    } else {
        // Edge tiles (2/32): unconditional clamped loads * {0,1} mask.
#pragma unroll 1
        for (int kk = 0; kk < KSTEPS; ++kk) {
            const int kbase = kk * 4 + half * 2;
            const int ca  = c0 + kbase - RAD;
            const int ca0 = min(max(ca,     0), NW - 1);
            const int ca1 = min(max(ca + 1, 0), NW - 1);
            const float m0 = ((unsigned)ca       < (unsigned)NW) ? 1.0f : 0.0f;
            const float m1 = ((unsigned)(ca + 1) < (unsigned)NW) ? 1.0f : 0.0f;
            v2f a;
            a.x = arowp[ca0] * m0;
            a.y = arowp[ca1] * m1;
            const v2f b = *(const v2f*)(band + kk * 64 + lane * 2);
            acc = __builtin_amdgcn_wmma_f32_16x16x4_f32(
                      false, a, false, b, (short)0, acc, false, false);
        }
    }
    // C/D layout: VGPR v -> row r0 + v + half*8, col c0 + lane%16
#pragma unroll
    for (int v = 0; v < 8; ++v) {
        q[(size_t)(r0 + v + half * 8) * NW + (c0 + l16)] = acc[v];
    }
}

// ---------------------------------------------------------------------------
// Vertical 21-tap conv fused with blending:
//   mp = vconv(tmp) * (1 - mask);  out_mp = mp;  out_init = mp + mask.
// C = A(16x36 banded weights) x B(36x16 input column slab).
// ---------------------------------------------------------------------------
__global__ __launch_bounds__(32)
void vconv_blend_wmma(const float* __restrict__ tmp, const float* __restrict__ mask,
                      float* __restrict__ out_init, float* __restrict__ out_mp,
                      const float* __restrict__ band) {
    const int lane  = threadIdx.x;
    const int tile  = blockIdx.x;
    const int plane = tile >> 10;
    const int ty    = (tile >> 5) & 31;
    const int tx    = tile & 31;
    const int r0 = ty * 16, c0 = tx * 16;

    const float* p = tmp + (size_t)plane * NH * NW;
    const int half = lane >> 4;
    const int l16  = lane & 15;
    const float* colp = p + (c0 + l16);    // B-matrix column for lane

    v8f acc = {};
    if (ty >= 1 && ty <= 30) {
        // Interior: rows r0-10 .. r0+25 all in-range.
#pragma unroll
        for (int kk = 0; kk < KSTEPS; ++kk) {
            const int kbase = kk * 4 + half * 2;
            const int rb = r0 + kbase - RAD;
            const v2f a = *(const v2f*)(band + kk * 64 + lane * 2);
            v2f b;
            b.x = colp[(size_t)rb * NW];
            b.y = colp[(size_t)(rb + 1) * NW];
            acc = __builtin_amdgcn_wmma_f32_16x16x4_f32(
                      false, a, false, b, (short)0, acc, false, false);
        }
    } else {
        // Edge tiles (2/32): unconditional clamped loads * {0,1} mask.
#pragma unroll 1
        for (int kk = 0; kk < KSTEPS; ++kk) {
            const int kbase = kk * 4 + half * 2;
            const int rb  = r0 + kbase - RAD;
            const int rb0 = min(max(rb,     0), NH - 1);
            const int rb1 = min(max(rb + 1, 0), NH - 1);
            const float m0 = ((unsigned)rb       < (unsigned)NH) ? 1.0f : 0.0f;
            const float m1 = ((unsigned)(rb + 1) < (unsigned)NH) ? 1.0f : 0.0f;
            const v2f a = *(const v2f*)(band + kk * 64 + lane * 2);
            v2f b;
            b.x = colp[(size_t)rb0 * NW] * m0;
            b.y = colp[(size_t)rb1 * NW] * m1;
            acc = __builtin_amdgcn_wmma_f32_16x16x4_f32(
                      false, a, false, b, (short)0, acc, false, false);
        }
    }
#pragma unroll
    for (int v = 0; v < 8; ++v) {
        const size_t idx = (size_t)plane * NH * NW
                         + (size_t)(r0 + v + half * 8) * NW + (c0 + l16);
        const float m  = mask[idx];
        const float mp = acc[v] * (1.0f - m);
        out_mp[idx]   = mp;
        out_init[idx] = mp + m;
    }
}

// ---------------------------------------------------------------------------
// Stage 1 reduction: sum |x-y| * mp_pre/(mp+eps), float4-vectorized.
// A 16B group never crosses a row/channel boundary (W, HW, CHW all % 4 == 0).
// ---------------------------------------------------------------------------
__global__ __launch_bounds__(256)
void rsv_reduce1(const float4* __restrict__ x4, const float4* __restrict__ y4,
                 const float4* __restrict__ pre4, const float4* __restrict__ fin4,
                 float* __restrict__ partials) {
    const size_t NT4  = (size_t)NB * NC * NH * NW / 4;
    const size_t CHW4 = (size_t)NC * NH * NW / 4;
    const size_t HW4  = (size_t)NH * NW / 4;
    float s = 0.0f;
    for (size_t i = (size_t)blockIdx.x * blockDim.x + threadIdx.x; i < NT4;
         i += (size_t)gridDim.x * blockDim.x) {
        const size_t b  = i / CHW4;
        const size_t hw = (i % CHW4) % HW4;
        const size_t ri = b * HW4 + hw;
        const float4 xp = x4[i], yp = y4[i], rp = pre4[ri], rf = fin4[ri];
        s += fabsf(xp.x - yp.x) * (rp.x / (rf.x + 1e-8f));
        s += fabsf(xp.y - yp.y) * (rp.y / (rf.y + 1e-8f));
        s += fabsf(xp.z - yp.z) * (rp.z / (rf.z + 1e-8f));
        s += fabsf(xp.w - yp.w) * (rp.w / (rf.w + 1e-8f));
    }
    __shared__ float sm[256];
    sm[threadIdx.x] = s;
    __syncthreads();
    for (int st = 128; st > 0; st >>= 1) {
        if ((int)threadIdx.x < st) sm[threadIdx.x] += sm[threadIdx.x + st];
        __syncthreads();
    }
    if (threadIdx.x == 0) partials[blockIdx.x] = sm[0];
}

__global__ __launch_bounds__(256)
void rsv_reduce2(const float* __restrict__ partials, float* __restrict__ out, int n) {
    __shared__ float sm[256];
    float s = 0.0f;
    for (int i = threadIdx.x; i < n; i += 256) s += partials[i];
    sm[threadIdx.x] = s;
    __syncthreads();
    for (int st = 128; st > 0; st >>= 1) {
        if ((int)threadIdx.x < st) sm[threadIdx.x] += sm[threadIdx.x + st];
        __syncthreads();
    }
    if (threadIdx.x == 0) out[0] = sm[0] / (float)((size_t)NC * NH * NW);
}

// ---------------------------------------------------------------------------
extern "C" void kernel_launch(void* const* d_in, const int* in_sizes, int n_in,
                              void* d_out, int out_size, void* d_ws, size_t ws_size,
                              hipStream_t stream) {
    const float* x    = (const float*)d_in[0];
    const float* y    = (const float*)d_in[1];
    const float* mask = (const float*)d_in[2];
    float* out = (float*)d_out;

    const size_t N1 = (size_t)NB * NH * NW;     // 4,194,304 floats per plane set
    float* ws       = (float*)d_ws;
    float* tmpB     = ws;                       // hconv output
    float* initB    = ws + N1;                  // next-iteration input
    float* mpPre    = ws + 2 * N1;              // mp (iter 7's write survives)
    float* mpFin    = ws + 3 * N1;              // mp at iter 8
    float* partials = ws + 4 * N1;              // 1024 floats
    float* band     = ws + 4 * N1 + 1024;       // 576 floats, pre-swizzled weights

    init_band<<<1, 32, 0, stream>>>(band);

    const dim3 blk(32);      // one wave per block, one 16x16 tile per wave
    const dim3 grd(16384);   // 16 planes * 32 * 32 tiles

    for (int it = 0; it < 9; ++it) {
        const float* src = (it == 0) ? mask : initB;
        hconv_wmma<<<grd, blk, 0, stream>>>(src, tmpB, band);
        float* mpDst = (it == 8) ? mpFin : mpPre;
        vconv_blend_wmma<<<grd, blk, 0, stream>>>(tmpB, mask, initB, mpDst, band);
    }

    rsv_reduce1<<<1024, 256, 0, stream>>>((const float4*)x, (const float4*)y,
                                          (const float4*)mpPre, (const float4*)mpFin,
                                          partials);
    rsv_reduce2<<<1, 256, 0, stream>>>(partials, out, 1024);
}